// InferHeight_12601434046716
// MI455X (gfx1250) — compile-verified
//
#include <hip/hip_runtime.h>
#include <math.h>
#include <stdint.h>

#define THREADS 256
#define WAVE    32
#define DEPTH   4      // LDS ring-buffer stages (3 prefetch stages in flight)

// ---- CDNA5 async global->LDS data mover (inline asm; portable across toolchains) ----
// GVS mode: mem_addr = SGPR64_base + VGPR32_byte_offset. First operand = LDS byte addr VGPR.
// Data is streamed exactly once (256 MiB > 192 MB L2) -> non-temporal load hint.
__device__ __forceinline__ void async_copy_b128(uint32_t lds_addr, uint32_t voffset, uint64_t sbase) {
  asm volatile("global_load_async_to_lds_b128 %0, %1, %2 th:TH_LOAD_NT"
               :: "v"(lds_addr), "v"(voffset), "s"(sbase)
               : "memory");
}
__device__ __forceinline__ void s_wait_async_le(int n) {  // n resolved at compile time when unrolled
  switch (n) {
    case 0: asm volatile("s_wait_asynccnt 0" ::: "memory"); break;
    case 2: asm volatile("s_wait_asynccnt 2" ::: "memory"); break;
    case 4: asm volatile("s_wait_asynccnt 4" ::: "memory"); break;
    default: asm volatile("s_wait_asynccnt 6" ::: "memory"); break;
  }
}
__device__ __forceinline__ void s_wait_ds0() { asm volatile("s_wait_dscnt 0" ::: "memory"); }

__device__ __forceinline__ uint32_t lds_addr_of(const void* p) {
  // Generic LDS pointers carry the LDS byte offset in the low 32 bits (aperture in high bits).
  return (uint32_t)(uintptr_t)p;
}

// norm of the xz components of lonlat2xyz(lon,lat) * (sgn / sin(lat))
__device__ __forceinline__ float ring_norm(float lon, float lat, float sgn) {
  float sl, cl, sp, cp;
  __sincosf(lon, &sl, &cl);   // native v_sin/v_cos; |lon| <= pi, |lat| <= 1.3
  __sincosf(lat, &sp, &cp);
  float inv = sgn / sp;       // sgn = -1 (up), +1 (down); sin(lat) bounded away from 0
  float x = cp * sl * inv;
  float z = cp * cl * inv;
  return sqrtf(x * x + z * z);
}

__device__ __forceinline__ void accum_pair(float4 u4, float4 d4, float& num, float& den) {
  {
    float un = ring_norm(u4.x, u4.y, -1.0f);
    float dv = ring_norm(d4.x, d4.y,  1.0f);
    num = fmaf(un, dv, num);
    den = fmaf(un, un, den);
  }
  {
    float un = ring_norm(u4.z, u4.w, -1.0f);
    float dv = ring_norm(d4.z, d4.w,  1.0f);
    num = fmaf(un, dv, num);
    den = fmaf(un, un, den);
  }
}

__device__ __forceinline__ void block_reduce_and_store(float num, float den, float* out, int b, int t,
                                                       float* rnum, float* rden) {
  for (int off = WAVE / 2; off > 0; off >>= 1) {
    num += __shfl_down(num, off, WAVE);
    den += __shfl_down(den, off, WAVE);
  }
  if ((t & (WAVE - 1)) == 0) {
    rnum[t / WAVE] = num;
    rden[t / WAVE] = den;
  }
  __syncthreads();
  if (t == 0) {
    float n = 0.0f, d = 0.0f;
#pragma unroll
    for (int w = 0; w < THREADS / WAVE; ++w) { n += rnum[w]; d += rden[w]; }
    out[b] = n / d;
  }
}

// ---------------- fully-unrolled specialization (N known => ITERS=8) ----------------
template <int ITERS>
__global__ __launch_bounds__(THREADS)
void infer_height_ratio_spec(const float* __restrict__ up,
                             const float* __restrict__ dn,
                             float* __restrict__ out,
                             int quads_per_row)
{
  __shared__ float4 tile[DEPTH][2][THREADS];   // 4*2*256*16 B = 32 KB
  __shared__ float  rnum[THREADS / WAVE];
  __shared__ float  rden[THREADS / WAVE];

  const int b = blockIdx.x;
  const int t = threadIdx.x;

  const uint64_t up_row = (uint64_t)(uintptr_t)(up + (size_t)b * (size_t)quads_per_row * 4);
  const uint64_t dn_row = (uint64_t)(uintptr_t)(dn + (size_t)b * (size_t)quads_per_row * 4);

  float num = 0.0f, den = 0.0f;

  // Prologue: fill up to DEPTH-1 = 3 stages of the ring.
#pragma unroll
  for (int s = 0; s < DEPTH - 1; ++s) {
    if (s < ITERS) {
      uint32_t voff = (uint32_t)(s * THREADS + t) * 16u;
      async_copy_b128(lds_addr_of(&tile[s][0][t]), voff, up_row);
      async_copy_b128(lds_addr_of(&tile[s][1][t]), voff, dn_row);
    }
  }

#pragma unroll
  for (int it = 0; it < ITERS; ++it) {
    const int buf = it & (DEPTH - 1);

    if (it + DEPTH - 1 < ITERS) {
      // The buffer being refilled was last read at iteration it-1: drain DS reads first
      // (async engine writes are unordered vs DS ops; ds_load done => data already in VGPRs).
      s_wait_ds0();
      uint32_t voff = (uint32_t)((it + DEPTH - 1) * THREADS + t) * 16u;
      async_copy_b128(lds_addr_of(&tile[(it + DEPTH - 1) & (DEPTH - 1)][0][t]), voff, up_row);
      async_copy_b128(lds_addr_of(&tile[(it + DEPTH - 1) & (DEPTH - 1)][1][t]), voff, dn_row);
    }

    // Async loads complete in order per wave: allow 2 ops per stage still in flight.
    const int rem = ITERS - 1 - it;                       // compile-time under full unroll
    s_wait_async_le(rem >= DEPTH - 1 ? 2 * (DEPTH - 1) : 2 * rem);

    float4 u4 = tile[buf][0][t];                          // ds_load_b128
    float4 d4 = tile[buf][1][t];
    accum_pair(u4, d4, num, den);
  }

  block_reduce_and_store(num, den, out, b, t, rnum, rden);
}

// ---------------- generic runtime-loop fallback ----------------
__global__ __launch_bounds__(THREADS)
void infer_height_ratio_generic(const float* __restrict__ up,
                                const float* __restrict__ dn,
                                float* __restrict__ out,
                                int quads_per_row, int iters)
{
  __shared__ float4 tile[DEPTH][2][THREADS];
  __shared__ float  rnum[THREADS / WAVE];
  __shared__ float  rden[THREADS / WAVE];

  const int b = blockIdx.x;
  const int t = threadIdx.x;

  const uint64_t up_row = (uint64_t)(uintptr_t)(up + (size_t)b * (size_t)quads_per_row * 4);
  const uint64_t dn_row = (uint64_t)(uintptr_t)(dn + (size_t)b * (size_t)quads_per_row * 4);

  float num = 0.0f, den = 0.0f;

  for (int s = 0; s < DEPTH - 1 && s < iters; ++s) {
    uint32_t voff = (uint32_t)(s * THREADS + t) * 16u;
    async_copy_b128(lds_addr_of(&tile[s][0][t]), voff, up_row);
    async_copy_b128(lds_addr_of(&tile[s][1][t]), voff, dn_row);
  }

  for (int it = 0; it < iters; ++it) {
    const int buf = it & (DEPTH - 1);

    if (it + DEPTH - 1 < iters) {
      s_wait_ds0();
      uint32_t voff = (uint32_t)((it + DEPTH - 1) * THREADS + t) * 16u;
      async_copy_b128(lds_addr_of(&tile[(it + DEPTH - 1) & (DEPTH - 1)][0][t]), voff, up_row);
      async_copy_b128(lds_addr_of(&tile[(it + DEPTH - 1) & (DEPTH - 1)][1][t]), voff, dn_row);
    }

    const int rem = iters - 1 - it;
    s_wait_async_le(rem >= DEPTH - 1 ? 2 * (DEPTH - 1) : 2 * rem);

    float4 u4 = tile[buf][0][t];
    float4 d4 = tile[buf][1][t];
    accum_pair(u4, d4, num, den);
  }

  block_reduce_and_store(num, den, out, b, t, rnum, rden);
}

extern "C" void kernel_launch(void* const* d_in, const int* in_sizes, int n_in,
                              void* d_out, int out_size, void* d_ws, size_t ws_size,
                              hipStream_t stream) {
  const float* up = (const float*)d_in[0];   // pred_up  (B, N, 2) fp32
  const float* dn = (const float*)d_in[1];   // pred_down(B, N, 2) fp32
  float* out = (float*)d_out;                // (B,) fp32

  const int B = out_size;                                   // 4096
  const long long tot = (long long)in_sizes[0];             // B*N*2
  const int N = (int)(tot / ((long long)B * 2));            // 4096
  const int quads = N / 2;                                  // 2048 float4 per row per input
  const int iters = quads / THREADS;                        // 8 pipeline stages

  if (iters == 8 && quads == iters * THREADS) {
    infer_height_ratio_spec<8><<<B, THREADS, 0, stream>>>(up, dn, out, quads);
  } else {
    infer_height_ratio_generic<<<B, THREADS, 0, stream>>>(up, dn, out, quads, iters);
  }
}